// GatedGINLayer_64682207478383
// MI455X (gfx1250) — compile-verified
//
#include <hip/hip_runtime.h>
#include <hip/hip_bf16.h>

#define N_NODES 100000
#define N_EDGES 1600000
#define CH 64

// LDS row strides (floats) chosen for bank-conflict avoidance on 64x4B banks:
//  - H tiles: stride 65 -> lane l hits bank (l + const) % 64, conflict-free.
//  - Weights: stride 72 -> the two half-wave K-row groups land on disjoint
//    16-bank groups (2 rows * 72 = 144 ≡ 16 mod 64).
#define HS 65
#define WS 72

typedef __attribute__((ext_vector_type(2))) float v2f;
typedef __attribute__((ext_vector_type(8))) float v8f;

// ---------------------------------------------------------------- zero agg
__global__ void gin_zero_agg(float4* __restrict__ agg4) {
  unsigned i = blockIdx.x * blockDim.x + threadIdx.x;   // N*CH/4 = 1.6M exact
  float4 z = {0.f, 0.f, 0.f, 0.f};
  agg4[i] = z;
}

// ------------------------------------------------- edge scatter-add (atomics)
// One thread per (edge, 4-channel quad): float4 gather of x[src] (L2-resident,
// x is 25.6MB << 192MB L2) + 4 global_atomic_add_f32 into agg[dst].
__global__ void gin_scatter_add(const float* __restrict__ x,
                                const long long* __restrict__ src_idx,
                                const long long* __restrict__ dst_idx,
                                float* __restrict__ agg) {
  unsigned t = blockIdx.x * blockDim.x + threadIdx.x;   // E*16 = 25.6M exact
  unsigned e = t >> 4;
  unsigned q = (t & 15u) << 2;                          // channel quad base
  long long s = src_idx[e];
  long long d = dst_idx[e];
  const float4 v = *(const float4*)(x + (size_t)s * CH + q);
  float* p = agg + (size_t)d * CH + q;
  atomicAdd(p + 0, v.x);
  atomicAdd(p + 1, v.y);
  atomicAdd(p + 2, v.z);
  atomicAdd(p + 3, v.w);
}

// --------------------------------------- fused (x+agg) -> MLP -> gate, WMMA
// 4 wave32 per block, 16-node tile per wave, 64 nodes per block.
// Each 16x16 output tile: K=64 as 16 steps of V_WMMA_F32_16X16X4_F32.
__global__ __launch_bounds__(128) void gin_mlp_wmma(
    const float* __restrict__ x, const float* __restrict__ agg,
    const float* __restrict__ W1, const float* __restrict__ b1,
    const float* __restrict__ W2, const float* __restrict__ b2,
    const float* __restrict__ alpha, float* __restrict__ out) {
  __shared__ float w1s[CH * WS];
  __shared__ float w2s[CH * WS];
  __shared__ float b1s[CH];
  __shared__ float b2s[CH];
  __shared__ float hs[4 * 16 * HS];   // per-wave 16x64 tile (h, then mid)

  const int tid  = threadIdx.x;
  const int lane = tid & 31;
  const int wave = tid >> 5;
  const int lrow = lane & 15;         // M (A rows) or N (B cols) index
  const int lhi  = lane >> 4;         // half-wave select: K offset {0,2}

  // Stage weights (row-padded). W stored [in,out] = [K,N] row-major.
  for (int i = tid; i < CH * CH; i += 128) {
    int r = i >> 6, c = i & 63;
    w1s[r * WS + c] = W1[i];
    w2s[r * WS + c] = W2[i];
  }
  if (tid < CH) { b1s[tid] = b1[tid]; b2s[tid] = b2[tid]; }

  // Stage h = x + agg for this wave's 16-node tile (coalesced over channels).
  const int tileBase = blockIdx.x * 64 + wave * 16;
  float* hw = hs + wave * (16 * HS);
  for (int i = lane; i < 16 * CH; i += 32) {
    int r = i >> 6, c = i & 63;
    int node = tileBase + r;
    if (node >= N_NODES) node = N_NODES - 1;   // clamp; rows never stored
    size_t off = (size_t)node * CH + c;
    hw[r * HS + c] = x[off] + agg[off];
  }
  __syncthreads();

  // ---- GEMM1: mid = relu(h @ W1 + b1), 4 N-tiles kept in registers.
  v8f mid[4];
  #pragma unroll
  for (int nt = 0; nt < 4; ++nt) {
    v8f c = {};
    #pragma unroll
    for (int kk = 0; kk < 16; ++kk) {
      const int ka = kk * 4 + lhi * 2;          // K = {0,1}|{2,3} per half-wave
      v2f a, b;
      a.x = hw[lrow * HS + ka];
      a.y = hw[lrow * HS + ka + 1];
      b.x = w1s[ka * WS + nt * 16 + lrow];
      b.y = w1s[(ka + 1) * WS + nt * 16 + lrow];
      c = __builtin_amdgcn_wmma_f32_16x16x4_f32(false, a, false, b,
                                                (short)0, c, false, false);
    }
    const float bias = b1s[nt * 16 + lrow];
    #pragma unroll
    for (int v = 0; v < 8; ++v) {
      float t = c[v] + bias;
      c[v] = t > 0.f ? t : 0.f;
    }
    mid[nt] = c;
  }
  __syncthreads();                 // everyone done reading h
  // Write mid tile back over the h tile (C/D layout: row = v + lhi*8).
  #pragma unroll
  for (int nt = 0; nt < 4; ++nt) {
    #pragma unroll
    for (int v = 0; v < 8; ++v)
      hw[(v + lhi * 8) * HS + nt * 16 + lrow] = mid[nt][v];
  }
  __syncthreads();

  // ---- GEMM2: out = gate * (mid @ W2 + b2), streamed straight to global.
  const float gate = 1.0f / (1.0f + __expf(-alpha[0]));
  #pragma unroll
  for (int nt = 0; nt < 4; ++nt) {
    v8f c = {};
    #pragma unroll
    for (int kk = 0; kk < 16; ++kk) {
      const int ka = kk * 4 + lhi * 2;
      v2f a, b;
      a.x = hw[lrow * HS + ka];
      a.y = hw[lrow * HS + ka + 1];
      b.x = w2s[ka * WS + nt * 16 + lrow];
      b.y = w2s[(ka + 1) * WS + nt * 16 + lrow];
      c = __builtin_amdgcn_wmma_f32_16x16x4_f32(false, a, false, b,
                                                (short)0, c, false, false);
    }
    const float bias = b2s[nt * 16 + lrow];
    #pragma unroll
    for (int v = 0; v < 8; ++v) {
      int node = tileBase + v + lhi * 8;
      if (node < N_NODES)
        out[(size_t)node * CH + nt * 16 + lrow] = gate * (c[v] + bias);
    }
  }

  if (blockIdx.x == 0 && tid == 0)
    out[(size_t)N_NODES * CH] = gate;     // trailing scalar output: gate
}

extern "C" void kernel_launch(void* const* d_in, const int* in_sizes, int n_in,
                              void* d_out, int out_size, void* d_ws, size_t ws_size,
                              hipStream_t stream) {
  (void)in_sizes; (void)n_in; (void)out_size; (void)ws_size;
  const float*     x     = (const float*)d_in[0];
  const long long* ei    = (const long long*)d_in[1];   // int64 in reference
  const float*     W1    = (const float*)d_in[2];
  const float*     b1    = (const float*)d_in[3];
  const float*     W2    = (const float*)d_in[4];
  const float*     b2    = (const float*)d_in[5];
  const float*     alpha = (const float*)d_in[6];
  float*           out   = (float*)d_out;
  float*           agg   = (float*)d_ws;                // N*CH fp32 = 25.6 MB

  // 1) zero the aggregation buffer: N*CH/4 float4 = 1,600,000 threads
  gin_zero_agg<<<(N_NODES * CH / 4) / 256, 256, 0, stream>>>((float4*)agg);

  // 2) edge scatter-add: E*16 threads = 25,600,000 (exact multiple of 256)
  gin_scatter_add<<<(N_EDGES * 16) / 256, 256, 0, stream>>>(
      x, ei, ei + N_EDGES, agg);

  // 3) fused GIN MLP: 64 nodes/block -> ceil(100000/64) = 1563 blocks
  gin_mlp_wmma<<<(N_NODES + 63) / 64, 128, 0, stream>>>(
      x, agg, W1, b1, W2, b2, alpha, out);
}